// SpectralPooling_65103114273312
// MI455X (gfx1250) — compile-verified
//
#include <hip/hip_runtime.h>
#include <hip/hip_bf16.h>

// Spectral pooling = 3x (48x64 matrix applied along one axis).
// P = dct_mat(48)^T[:, :32] @ dct_mat(64)[:32, :]   (forward DCT + truncate + pad + iDCT fused)

typedef __attribute__((ext_vector_type(2))) float v2f;
typedef __attribute__((ext_vector_type(4))) float v4f;
typedef __attribute__((ext_vector_type(8))) float v8f;

#define PI_F 3.14159265358979323846f

// ---------------------------------------------------------------------------
// Build the fused 48x64 operator P in workspace.
// P[a][d] = sum_{k=0}^{31} M48[k][a] * M64[k][d]
// ---------------------------------------------------------------------------
__global__ void build_P_kernel(float* __restrict__ P) {
    int idx = blockIdx.x * blockDim.x + threadIdx.x;
    if (idx >= 48 * 64) return;
    int a = idx / 64;
    int d = idx % 64;
    float s = 0.0f;
    const float s48 = sqrtf(2.0f / 48.0f);
    const float s64 = sqrtf(2.0f / 64.0f);
    for (int k = 0; k < 32; ++k) {
        float r = (k == 0) ? 0.70710678118654752f : 1.0f; // 1/sqrt(2) on DC row
        float f48 = cosf(PI_F * (a + 0.5f) * (float)k / 48.0f) * s48 * r;
        float f64 = cosf(PI_F * (d + 0.5f) * (float)k / 64.0f) * s64 * r;
        s += f48 * f64;
    }
    P[idx] = s;
}

// ---------------------------------------------------------------------------
// Generic stage: Y[(outer*48 + a)*inner + innerRow] = sum_k X[r*64+k] * P[a*64+k]
// with r = outer*inner + innerRow.  R rows, K = 64, A = 48.
// One wave computes a 16-row x 48-col tile via 3 f32 WMMA accumulators.
// blockDim = 256 (8 waves) -> 128 rows per block.  inner % 16 == 0 guaranteed.
// ---------------------------------------------------------------------------
__global__ __launch_bounds__(256) void dct_stage_kernel(
    const float* __restrict__ X, const float* __restrict__ P,
    float* __restrict__ Y, int innerSize) {

    __shared__ float sP[48 * 64];
    // Cooperative stage of the 12KB operator into LDS (shared by all waves).
    for (int i = threadIdx.x; i < 48 * 64; i += 256) sP[i] = P[i];
    __syncthreads();

    const int wave  = threadIdx.x >> 5;
    const int lane  = threadIdx.x & 31;
    const int l16   = lane & 15;       // M row / N col within tile
    const int lhalf = lane >> 4;       // selects K pair (A/B frag) and row half (C)

    const int rowTile = blockIdx.x * 128 + wave * 16;   // first row of this wave's tile
    const float* xrow = X + (size_t)(rowTile + l16) * 64;

    v8f c0 = {}, c1 = {}, c2 = {};

    #pragma unroll
    for (int k = 0; k < 64; k += 4) {
        const int k0 = k + 2 * lhalf;
        // A fragment (16x4 f32): lane l16 = row M, VGPR0/1 = K=k0,k0+1
        v2f a  = *(const v2f*)(xrow + k0);
        // B fragments (4x16 f32): lane l16 = col N (a-index), VGPR0/1 = K=k0,k0+1
        v2f b0 = *(const v2f*)(sP + (0 * 16 + l16) * 64 + k0);
        v2f b1 = *(const v2f*)(sP + (1 * 16 + l16) * 64 + k0);
        v2f b2 = *(const v2f*)(sP + (2 * 16 + l16) * 64 + k0);
        c0 = __builtin_amdgcn_wmma_f32_16x16x4_f32(false, a, false, b0, (short)0, c0, false, false);
        c1 = __builtin_amdgcn_wmma_f32_16x16x4_f32(false, a, false, b1, (short)0, c1, false, false);
        c2 = __builtin_amdgcn_wmma_f32_16x16x4_f32(false, a, false, b2, (short)0, c2, false, false);
    }

    // Transposed store: out[(outer*48 + a)*inner + innerRow].
    // C layout: lane n holds column a=n; VGPR v = row v (lanes 0-15) or v+8 (16-31).
    // -> each lane's 8 accumulator values land at 8 consecutive addresses.
    const int outer    = rowTile / innerSize;
    const int innerRow = (rowTile - outer * innerSize) + lhalf * 8;

    v8f acc[3] = {c0, c1, c2};
    #pragma unroll
    for (int t = 0; t < 3; ++t) {
        const int a = t * 16 + l16;
        float* dst = Y + (size_t)(outer * 48 + a) * innerSize + innerRow;
        v4f lo = __builtin_shufflevector(acc[t], acc[t], 0, 1, 2, 3);
        v4f hi = __builtin_shufflevector(acc[t], acc[t], 4, 5, 6, 7);
        *(v4f*)(dst)     = lo;   // global_store_b128
        *(v4f*)(dst + 4) = hi;   // global_store_b128
    }
}

// ---------------------------------------------------------------------------
// Host-side launch.  x: (4,32,64,64,64) f32 -> y: (4,32,48,48,48) f32.
// BC=128 processed in 4 chunks of 32 so scratch stays ~44MB and intermediates
// live in L2 (192MB).
// ---------------------------------------------------------------------------
extern "C" void kernel_launch(void* const* d_in, const int* in_sizes, int n_in,
                              void* d_out, int out_size, void* d_ws, size_t ws_size,
                              hipStream_t stream) {
    const float* x = (const float*)d_in[0];
    float* out = (float*)d_out;

    float* P   = (float*)d_ws;                 // 48*64 floats (pad to 4096)
    float* ws1 = P + 4096;                     // 32*48*64*64 = 6,291,456 floats
    float* ws2 = ws1 + (size_t)32 * 48 * 64 * 64; // 32*48*48*64 = 4,718,592 floats

    build_P_kernel<<<12, 256, 0, stream>>>(P);

    const size_t inVol  = (size_t)64 * 64 * 64;   // per (b,c) volume
    const size_t outVol = (size_t)48 * 48 * 48;

    for (int chunk = 0; chunk < 4; ++chunk) {
        const float* xin = x   + (size_t)chunk * 32 * inVol;
        float*      yout = out + (size_t)chunk * 32 * outVol;

        // Stage 1: contract W (rows = 32*64*64 = 131072), out [bc, a2, d, h]
        dct_stage_kernel<<<131072 / 128, 256, 0, stream>>>(xin, P, ws1, 64 * 64);
        // Stage 2: contract H (rows = 32*48*64 = 98304),  out [bc, a1, a2, d]
        dct_stage_kernel<<<98304 / 128, 256, 0, stream>>>(ws1, P, ws2, 48 * 64);
        // Stage 3: contract D (rows = 32*48*48 = 73728),  out [bc, a0, a1, a2]
        dct_stage_kernel<<<73728 / 128, 256, 0, stream>>>(ws2, P, yout, 48 * 48);
    }
}